// GCN_91250875171133
// MI455X (gfx1250) — compile-verified
//
#include <hip/hip_runtime.h>
#include <hip/hip_bf16.h>

typedef float v2f __attribute__((ext_vector_type(2)));
typedef float v8f __attribute__((ext_vector_type(8)));

#define TILE_N     32
#define LDS_STRIDE 260   // 256 + 4 pad: lane n hits bank (4n + k) -> conflict-free b64 reads

// ---------------------------------------------------------------------------
// C[M x Kout] = A[M x K] * W[K x Kout]   (fp32 exact, V_WMMA_F32_16X16X4_F32)
// Block: 256 threads (8 waves), covers 256 rows x 32 cols.
//   - W slice [K x 32] staged transposed in LDS once per block (33 KB).
//   - Each wave: 32x32 output tile = 2x2 WMMA tiles, 4 accumulators.
//   - Inner loop per k-step: 2 global b64 (A) + 2 ds b64 (B, immediate offsets)
//     feeding 4 WMMAs  ->  1 load per 2048-FLOP WMMA.
// Requires: K == 256 (== blockDim), Kout % 32 == 0, M % 16 == 0.
// ---------------------------------------------------------------------------
__global__ void __launch_bounds__(256)
gemm_wmma_f32(const float* __restrict__ A, const float* __restrict__ W,
              float* __restrict__ C, int M, int K, int Kout) {
    __shared__ float wt[TILE_N * LDS_STRIDE];   // wt[n*LDS_STRIDE + k] = W[k][n0+n]

    const int tid  = threadIdx.x;
    const int lane = tid & 31;
    const int wave = tid >> 5;
    const int n0   = blockIdx.y * TILE_N;

    // ---- cooperative transpose-fill of W slice (thread t owns row k = t) ----
    for (int k = tid; k < K; k += 256) {
        const float* wrow = W + (size_t)k * Kout + n0;
#pragma unroll
        for (int n = 0; n < TILE_N; n += 4) {
            float4 w4 = *(const float4*)(wrow + n);
            wt[(n + 0) * LDS_STRIDE + k] = w4.x;
            wt[(n + 1) * LDS_STRIDE + k] = w4.y;
            wt[(n + 2) * LDS_STRIDE + k] = w4.z;
            wt[(n + 3) * LDS_STRIDE + k] = w4.w;
        }
    }
    __syncthreads();

    // ---- wave tile ----------------------------------------------------------
    const int m0 = (blockIdx.x * 8 + wave) * 32;
    if (m0 >= M) return;                        // wave-uniform: EXEC stays all-1s
    const int  half  = lane >> 4;               // selects K pair {2h, 2h+1}
    const int  l15   = lane & 15;
    const bool hasM1 = (m0 + 16) < M;

    int r1 = m0 + 16 + l15;
    if (r1 >= M) r1 = M - 1;                    // clamp: compute garbage, skip store
    const float* arow0 = A + (size_t)(m0 + l15) * K + 2 * half;
    const float* arow1 = A + (size_t)r1 * K + 2 * half;

    const int bb0 = l15 * LDS_STRIDE + 2 * half;
    const int bb1 = bb0 + 16 * LDS_STRIDE;

    v8f acc00 = {}, acc01 = {}, acc10 = {}, acc11 = {};

    for (int k = 0; k < K; k += 4) {
        v2f a0 = *(const v2f*)(arow0 + k);
        v2f a1 = *(const v2f*)(arow1 + k);
        v2f b0 = *(const v2f*)&wt[bb0 + k];     // ds_load_b64, immediate offset
        v2f b1 = *(const v2f*)&wt[bb1 + k];

        acc00 = __builtin_amdgcn_wmma_f32_16x16x4_f32(false, a0, false, b0,
                                                      (short)0, acc00, false, false);
        acc01 = __builtin_amdgcn_wmma_f32_16x16x4_f32(false, a0, false, b1,
                                                      (short)0, acc01, false, false);
        acc10 = __builtin_amdgcn_wmma_f32_16x16x4_f32(false, a1, false, b0,
                                                      (short)0, acc10, false, false);
        acc11 = __builtin_amdgcn_wmma_f32_16x16x4_f32(false, a1, false, b1,
                                                      (short)0, acc11, false, false);
    }

    // ---- store: C/D layout -> VGPR v: lanes 0-15 row m+v, lanes 16-31 row m+8+v
    float* crow0 = C + (size_t)(m0 + 8 * half) * Kout + n0 + l15;
#pragma unroll
    for (int v = 0; v < 8; ++v) {
        crow0[(size_t)v * Kout]      = acc00[v];
        crow0[(size_t)v * Kout + 16] = acc01[v];
    }
    if (hasM1) {
        float* crow1 = crow0 + (size_t)16 * Kout;
#pragma unroll
        for (int v = 0; v < 8; ++v) {
            crow1[(size_t)v * Kout]      = acc10[v];
            crow1[(size_t)v * Kout + 16] = acc11[v];
        }
    }
}

// ---------------------------------------------------------------------------
// out[rows[e]] += vals[e] * t[cols[e]]  : one wave per edge, lane = float4 slice.
// t working set (<= 51.2 MB) sits in the 192 MB L2 -> gathers hit L2; scatter
// uses hardware fp32 atomic adds absorbed by L2 atomic units.
// ---------------------------------------------------------------------------
template <int F>
__global__ void __launch_bounds__(256)
spmm_scatter(const float* __restrict__ t, const float* __restrict__ vals,
             const int* __restrict__ rows, const int* __restrict__ cols,
             float* __restrict__ out, int nEdges) {
    const int edge = blockIdx.x * 8 + (threadIdx.x >> 5);
    if (edge >= nEdges) return;
    const int lane = threadIdx.x & 31;

    const int   r = rows[edge];
    const int   c = cols[edge];
    const float v = vals[edge];

    const float* src = t   + (size_t)c * F;
    float*       dst = out + (size_t)r * F;

#pragma unroll
    for (int f = lane * 4; f < F; f += 128) {
        float4 hv = *(const float4*)(src + f);
        __hip_atomic_fetch_add(dst + f + 0, v * hv.x, __ATOMIC_RELAXED, __HIP_MEMORY_SCOPE_AGENT);
        __hip_atomic_fetch_add(dst + f + 1, v * hv.y, __ATOMIC_RELAXED, __HIP_MEMORY_SCOPE_AGENT);
        __hip_atomic_fetch_add(dst + f + 2, v * hv.z, __ATOMIC_RELAXED, __HIP_MEMORY_SCOPE_AGENT);
        __hip_atomic_fetch_add(dst + f + 3, v * hv.w, __ATOMIC_RELAXED, __HIP_MEMORY_SCOPE_AGENT);
    }
}

// ---------------------------------------------------------------------------
// h[i] = act(h[i] + b[i & Fmask]) ; F power of two (256 / 128).
// ---------------------------------------------------------------------------
__global__ void __launch_bounds__(256)
bias_act(float* __restrict__ h, const float* __restrict__ b, int total, int Fmask, int relu) {
    const int i = blockIdx.x * blockDim.x + threadIdx.x;
    if (i >= total) return;
    float x = h[i] + b[i & Fmask];
    h[i] = relu ? fmaxf(x, 0.0f) : x;
}

extern "C" void kernel_launch(void* const* d_in, const int* in_sizes, int n_in,
                              void* d_out, int out_size, void* d_ws, size_t ws_size,
                              hipStream_t stream) {
    const float* x    = (const float*)d_in[0];
    const float* vals = (const float*)d_in[1];
    const int*   rows = (const int*)  d_in[2];
    const int*   cols = (const int*)  d_in[3];
    const float* W1   = (const float*)d_in[4];
    const float* b1   = (const float*)d_in[5];
    const float* W2   = (const float*)d_in[6];
    const float* b2   = (const float*)d_in[7];
    const float* W3   = (const float*)d_in[8];
    const float* b3   = (const float*)d_in[9];

    const int N = in_sizes[0] / 256;   // 50000
    const int E = in_sizes[1];         // 800000

    float* bufA = (float*)d_ws;                 // N*256 floats
    float* bufB = bufA + (size_t)N * 256;       // N*256 floats
    float* outF = (float*)d_out;                // N*128 floats

    const dim3 blk(256);
    // 32-row wave tiles: ceil(N/32) of them, 8 waves per block
    const int  mTiles  = (N + 31) / 32;
    const int  gBlocks = (mTiles + 7) / 8;
    const int  eBlocks = (E + 7) / 8;

    // ---- Layer 1: t = x @ W1 ; s = spmm(t) ; h1 = relu(s + b1) --------------
    gemm_wmma_f32<<<dim3(gBlocks, 256 / 32), blk, 0, stream>>>(x, W1, bufA, N, 256, 256);
    (void)hipMemsetAsync(bufB, 0, (size_t)N * 256 * sizeof(float), stream);
    spmm_scatter<256><<<dim3(eBlocks), blk, 0, stream>>>(bufA, vals, rows, cols, bufB, E);
    bias_act<<<dim3((N * 256 + 255) / 256), blk, 0, stream>>>(bufB, b1, N * 256, 255, 1);

    // ---- Layer 2: t = h1 @ W2 ; s = spmm(t) ; h2 = relu(s + b2) -------------
    gemm_wmma_f32<<<dim3(gBlocks, 256 / 32), blk, 0, stream>>>(bufB, W2, bufA, N, 256, 256);
    (void)hipMemsetAsync(bufB, 0, (size_t)N * 256 * sizeof(float), stream);
    spmm_scatter<256><<<dim3(eBlocks), blk, 0, stream>>>(bufA, vals, rows, cols, bufB, E);
    bias_act<<<dim3((N * 256 + 255) / 256), blk, 0, stream>>>(bufB, b2, N * 256, 255, 1);

    // ---- Layer 3: t = h2 @ W3 ; out = spmm(t) + b3 --------------------------
    gemm_wmma_f32<<<dim3(gBlocks, 128 / 32), blk, 0, stream>>>(bufB, W3, bufA, N, 256, 128);
    (void)hipMemsetAsync(outF, 0, (size_t)N * 128 * sizeof(float), stream);
    spmm_scatter<128><<<dim3(eBlocks), blk, 0, stream>>>(bufA, vals, rows, cols, outF, E);
    bias_act<<<dim3((N * 128 + 255) / 256), blk, 0, stream>>>(outF, b3, N * 128, 127, 0);
}